// ThousandsBrainsNet_82179904242263
// MI455X (gfx1250) — compile-verified
//
#include <hip/hip_runtime.h>

typedef float v2f __attribute__((ext_vector_type(2)));
typedef float v8f __attribute__((ext_vector_type(8)));

#define BSZ   256
#define HIMG  224
#define WIMG  224
#define NPOS  73          // (224-7)/3 + 1
#define PLD   64          // pmean leading dim (49 padded to 64)
#define HID   512
#define NCLS  10
#define KPAD  52          // 49 padded to multiple of 4

// workspace layout (float offsets)
#define WS_PMEAN 0
#define WS_W1EFF (WS_PMEAN + BSZ * PLD)        // 16384
#define WS_BIAS1 (WS_W1EFF + KPAD * HID)       // 43008
#define WS_W2EFF (WS_BIAS1 + HID)              // 43520
#define WS_H     (WS_W2EFF + HID * 16)         // 51712
// total = 51712 + 256*512 = 182784 floats (~715 KB)

// ---------------------------------------------------------------------------
// Kernel 1: pmean[b][fh*7+fw] = mean over 73x73 positions of x[b, 3i+fh, 3j+fw]
// block = (b, fh); 7 waves, wave w handles fw=w; lane handles j in {l, l+32, l+64}
// ---------------------------------------------------------------------------
__global__ void pmean_kernel(const float* __restrict__ x, float* __restrict__ ws) {
  const int b    = blockIdx.x / 7;
  const int fh   = blockIdx.x % 7;
  const int fw   = threadIdx.x >> 5;   // wave id 0..6
  const int lane = threadIdx.x & 31;
  const float* img = x + (size_t)b * (HIMG * WIMG);
  const int c1 = 3 * lane + fw;        // col for j = lane
  float acc = 0.0f;
  #pragma unroll 4
  for (int i = 0; i < NPOS; ++i) {
    const float* row = img + (3 * i + fh) * WIMG;
    acc += row[c1];                    // j = lane        (always < 73)
    acc += row[c1 + 96];               // j = lane + 32   (always < 73)
    if (lane <= 8) acc += row[c1 + 192]; // j = lane + 64 (valid iff <= 72)
  }
  // wave32 reduction
  for (int off = 16; off; off >>= 1) acc += __shfl_down(acc, off, 32);
  if (lane == 0)
    ws[WS_PMEAN + b * PLD + fh * 7 + fw] = acc * (1.0f / (float)(NPOS * NPOS));
  // zero the K padding (cols 49..63) once per image
  if (fh == 0 && threadIdx.x < (PLD - 49))
    ws[WS_PMEAN + b * PLD + 49 + threadIdx.x] = 0.0f;
}

// ---------------------------------------------------------------------------
// Kernel 2: W1eff (KPADx512, rows>=49 zero), bias1eff (512), W2eff (512x16 padded)
// 2 blocks x 256 threads; thread owns output column o
// ---------------------------------------------------------------------------
__global__ void prep_kernel(const float* __restrict__ W, const float* __restrict__ bmc,
                            const float* __restrict__ W1, const float* __restrict__ b1,
                            const float* __restrict__ W2, float* __restrict__ ws) {
  __shared__ float sWm[16], sbm[16];
  if (threadIdx.x < 16) {
    float sw = 0.0f, sb = 0.0f;
    for (int m = 0; m < 64; ++m) {
      sw += W[m * 16 + threadIdx.x];
      sb += bmc[m * 16 + threadIdx.x];
    }
    sWm[threadIdx.x] = sw * (1.0f / 64.0f);
    sbm[threadIdx.x] = sb * (1.0f / 64.0f);
  }
  __syncthreads();
  const int o = blockIdx.x * blockDim.x + threadIdx.x;   // 0..511
  float bias = b1[o];
  for (int p = 0; p < 49; ++p) {
    float e = 0.0f;
    #pragma unroll
    for (int n = 0; n < 16; ++n) {
      const float w = W1[(n * 49 + p) * HID + o];        // coalesced over o
      e    += sWm[n] * w;
      bias += sbm[n] * w;
    }
    ws[WS_W1EFF + p * HID + o] = e;
  }
  for (int p = 49; p < KPAD; ++p) ws[WS_W1EFF + p * HID + o] = 0.0f;
  ws[WS_BIAS1 + o] = bias;
  #pragma unroll
  for (int c = 0; c < 16; ++c)
    ws[WS_W2EFF + o * 16 + c] = (c < NCLS) ? W2[o * NCLS + c] : 0.0f;
}

// ---------------------------------------------------------------------------
// Kernel 3: h = relu(pmean(256x49) @ W1eff(49x512) + bias1eff), fp32 WMMA 16x16x4
// 512 waves, one 16x16 tile each; K = 52 -> 13 WMMA steps
// ---------------------------------------------------------------------------
__global__ void fc1_wmma_kernel(float* __restrict__ ws) {
  const float* pmean = ws + WS_PMEAN;
  const float* w1e   = ws + WS_W1EFF;
  const float* bias1 = ws + WS_BIAS1;
  float* h           = ws + WS_H;
  const int wave = (blockIdx.x * blockDim.x + threadIdx.x) >> 5;  // 0..511
  const int lane = threadIdx.x & 31;
  const int tm = (wave >> 5) * 16;   // 16 m-tiles
  const int tn = (wave & 31) * 16;   // 32 n-tiles
  const int lh = lane >> 4;          // lane half: 0 -> K{0,1}, 1 -> K{2,3}
  const int ll = lane & 15;
  v8f c = {};
  #pragma unroll
  for (int s = 0; s < KPAD / 4; ++s) {
    const int k0 = 4 * s;
    v2f a, bb;
    // A 16x4 (ISA layout): lanes 0-15 row=ll VGPR0=K0,VGPR1=K1; lanes 16-31 K2,K3
    const float* ap = pmean + (tm + ll) * PLD + k0 + 2 * lh;
    a.x = ap[0]; a.y = ap[1];
    // B 4x16: row-of-B striped across lanes per VGPR; lane half selects K{0,1}/K{2,3}
    const float* bp = w1e + (k0 + 2 * lh) * HID + tn + ll;
    bb.x = bp[0]; bb.y = bp[HID];
    c = __builtin_amdgcn_wmma_f32_16x16x4_f32(false, a, false, bb,
                                              (short)0, c, false, false);
  }
  #pragma unroll
  for (int v = 0; v < 8; ++v) {
    const int row = tm + v + 8 * lh;   // C/D layout: VGPR v -> row v / v+8
    const int col = tn + ll;
    const float val = c[v] + bias1[col];
    h[row * HID + col] = fmaxf(val, 0.0f);
  }
}

// ---------------------------------------------------------------------------
// Kernel 4: out = h(256x512) @ W2eff(512x16) + b2, store cols 0..9
// 16 waves, K = 512 -> 128 WMMA steps
// ---------------------------------------------------------------------------
__global__ void fc2_wmma_kernel(const float* __restrict__ b2,
                                float* __restrict__ ws, float* __restrict__ out) {
  const float* h   = ws + WS_H;
  const float* w2e = ws + WS_W2EFF;
  const int wave = (blockIdx.x * blockDim.x + threadIdx.x) >> 5;  // 0..15
  const int lane = threadIdx.x & 31;
  const int tm = wave * 16;
  const int lh = lane >> 4;
  const int ll = lane & 15;
  v8f c = {};
  #pragma unroll 4
  for (int k0 = 0; k0 < HID; k0 += 4) {
    v2f a, bb;
    const float* ap = h + (tm + ll) * HID + k0 + 2 * lh;
    a.x = ap[0]; a.y = ap[1];
    const float* bp = w2e + (k0 + 2 * lh) * 16 + ll;
    bb.x = bp[0]; bb.y = bp[16];
    c = __builtin_amdgcn_wmma_f32_16x16x4_f32(false, a, false, bb,
                                              (short)0, c, false, false);
  }
  if (ll < NCLS) {                     // divergence only after all WMMAs
    #pragma unroll
    for (int v = 0; v < 8; ++v) {
      const int row = tm + v + 8 * lh;
      out[row * NCLS + ll] = c[v] + b2[ll];
    }
  }
}

// ---------------------------------------------------------------------------
extern "C" void kernel_launch(void* const* d_in, const int* in_sizes, int n_in,
                              void* d_out, int out_size, void* d_ws, size_t ws_size,
                              hipStream_t stream) {
  const float* x   = (const float*)d_in[0];  // (256,1,224,224)
  const float* W   = (const float*)d_in[1];  // (64,16)
  const float* bmc = (const float*)d_in[2];  // (64,16)
  const float* W1  = (const float*)d_in[3];  // (784,512)
  const float* b1  = (const float*)d_in[4];  // (512)
  const float* W2  = (const float*)d_in[5];  // (512,10)
  const float* b2  = (const float*)d_in[6];  // (10)
  float* ws  = (float*)d_ws;
  float* out = (float*)d_out;                // (256,10)

  pmean_kernel<<<BSZ * 7, 224, 0, stream>>>(x, ws);
  prep_kernel<<<2, 256, 0, stream>>>(W, bmc, W1, b1, W2, ws);
  fc1_wmma_kernel<<<128, 128, 0, stream>>>(ws);        // 512 waves
  fc2_wmma_kernel<<<4, 128, 0, stream>>>(b2, ws, out); // 16 waves
}